// PointTrans_Layer_down_90108413870726
// MI455X (gfx1250) — compile-verified
//
#include <hip/hip_runtime.h>
#include <hip/hip_bf16.h>

#define N_PTS   16384
#define C_IN    512
#define C_OUT   512
#define E_EDGES 262144
#define M_OUT   (N_PTS / 2)
#define BN_EPS  1e-5f

typedef float v2f __attribute__((ext_vector_type(2)));
typedef float v8f __attribute__((ext_vector_type(8)));

#define WPITCH 68  // LDS row pitch (floats): 16B-aligned rows, conflict-free frags

// ---------------------------------------------------------------------------
// GEMM: h = x @ W^T + b  via V_WMMA_F32_16X16X4_F32
// block = 256 threads (8 waves). Block tile 128 rows x 64 cols.
// Each wave: 16 rows x 64 cols = 4 accumulators of 16x16, A reused 4x.
// W slab (64 cols x 64 k = 16KB) double-buffered in LDS via
// global_load_async_to_lds_b128 (ASYNCcnt), shared by all 8 waves.
// A frag (16x4 f32): lane<16 -> M=lane, K=0,1 ; lane>=16 -> M=lane-16, K=2,3
// B frag (4x16 f32): B[k][n] = W[c0+n][k], read from LDS (ds_load_b64)
// C/D   (16x16 f32): VGPR r -> M = r + 8*(lane>=16), N = lane%16
// ---------------------------------------------------------------------------
__global__ __launch_bounds__(256) void gemm_wmma_kernel(
    const float* __restrict__ x, const float* __restrict__ W,
    const float* __restrict__ bias, float* __restrict__ h) {
  __shared__ float wtile[2][64 * WPITCH];

  const int wave = threadIdx.x >> 5;
  const int lane = threadIdx.x & 31;
  const int half = lane >> 4;
  const int l16  = lane & 15;
  const int r0 = blockIdx.x * 128 + wave * 16;
  const int c0 = blockIdx.y * 64;

  const float* arow = x + (size_t)(r0 + l16) * C_IN + 2 * half;

  // async-staging mapping: 4 threads per W row, 16 floats (4 x b128) each
  const int lr = threadIdx.x >> 2;        // 0..63: column index within tile
  const int lq = (threadIdx.x & 3) * 16;  // float offset within k-slab row

  auto issue_slab = [&](int ks, int buf) {
    const unsigned long long ga =
        (unsigned long long)(uintptr_t)(W + (size_t)(c0 + lr) * C_IN + ks + lq);
    const unsigned lds =
        (unsigned)(uintptr_t)&wtile[buf][lr * WPITCH + lq];
#pragma unroll
    for (int u = 0; u < 4; ++u) {
      asm volatile("global_load_async_to_lds_b128 %0, %1, off"
                   :: "v"(lds + 16u * u), "v"(ga + 16ull * u)
                   : "memory");
    }
  };

  v8f acc0 = {}; v8f acc1 = {}; v8f acc2 = {}; v8f acc3 = {};

  issue_slab(0, 0);

  for (int s = 0; s < C_IN / 64; ++s) {
    const int buf = s & 1;
    if (s < C_IN / 64 - 1) {
      issue_slab((s + 1) * 64, buf ^ 1);
      asm volatile("s_wait_asynccnt 0x4" ::: "memory");
    } else {
      asm volatile("s_wait_asynccnt 0x0" ::: "memory");
    }
    __syncthreads();  // slab s visible to all waves

    const float* wt = &wtile[buf][0];
    const int ks = s * 64;
#pragma unroll
    for (int kk = 0; kk < 64; kk += 4) {
      __builtin_prefetch(arow + ks + kk + 64, 0, 0);
      v2f a  = *(const v2f*)(arow + ks + kk);
      v2f b0 = *(const v2f*)(wt + (0  + l16) * WPITCH + kk + 2 * half);
      v2f b1 = *(const v2f*)(wt + (16 + l16) * WPITCH + kk + 2 * half);
      v2f b2 = *(const v2f*)(wt + (32 + l16) * WPITCH + kk + 2 * half);
      v2f b3 = *(const v2f*)(wt + (48 + l16) * WPITCH + kk + 2 * half);
      acc0 = __builtin_amdgcn_wmma_f32_16x16x4_f32(false, a, false, b0,
                                                   (short)0, acc0, false, false);
      acc1 = __builtin_amdgcn_wmma_f32_16x16x4_f32(false, a, false, b1,
                                                   (short)0, acc1, false, false);
      acc2 = __builtin_amdgcn_wmma_f32_16x16x4_f32(false, a, false, b2,
                                                   (short)0, acc2, false, false);
      acc3 = __builtin_amdgcn_wmma_f32_16x16x4_f32(false, a, false, b3,
                                                   (short)0, acc3, false, false);
    }
    __syncthreads();  // all waves done reading buf before it is overwritten
  }

  const float bb0 = bias[c0 + l16];
  const float bb1 = bias[c0 + 16 + l16];
  const float bb2 = bias[c0 + 32 + l16];
  const float bb3 = bias[c0 + 48 + l16];
#pragma unroll
  for (int r = 0; r < 8; ++r) {
    const int row = r0 + r + 8 * half;
    float* hp = h + (size_t)row * C_OUT + c0 + l16;
    hp[0]  = acc0[r] + bb0;
    hp[16] = acc1[r] + bb1;
    hp[32] = acc2[r] + bb2;
    hp[48] = acc3[r] + bb3;
  }
}

// ---------------------------------------------------------------------------
// Zero the accumulators / flags we need (d_ws is poisoned, not re-zeroed).
// ---------------------------------------------------------------------------
__global__ void init_kernel(float* sums, float* sumsq, int* selected) {
  const int i = blockIdx.x * blockDim.x + threadIdx.x;
  if (i < C_OUT) { sums[i] = 0.0f; sumsq[i] = 0.0f; }
  if (i < N_PTS) selected[i] = 0;
}

// ---------------------------------------------------------------------------
// BN stats: per-channel sum / sumsq. Each block reduces 64 rows, 2 ch/thread.
// ---------------------------------------------------------------------------
__global__ __launch_bounds__(256) void bn_stats_kernel(
    const float* __restrict__ h, float* __restrict__ sums,
    float* __restrict__ sumsq) {
  const int c2 = threadIdx.x * 2;
  const size_t base = (size_t)blockIdx.x * 64 * C_OUT;
  float s0 = 0.f, s1 = 0.f, q0 = 0.f, q1 = 0.f;
#pragma unroll 4
  for (int r = 0; r < 64; ++r) {
    float2 v = *(const float2*)(h + base + (size_t)r * C_OUT + c2);
    s0 += v.x; s1 += v.y;
    q0 += v.x * v.x; q1 += v.y * v.y;
  }
  atomicAdd(&sums[c2],      s0);
  atomicAdd(&sums[c2 + 1],  s1);
  atomicAdd(&sumsq[c2],     q0);
  atomicAdd(&sumsq[c2 + 1], q1);
}

// ---------------------------------------------------------------------------
// BN apply (training-mode batch stats) + ReLU, in place. float2 per thread.
// ---------------------------------------------------------------------------
__global__ __launch_bounds__(256) void bn_apply_kernel(
    float* __restrict__ h, const float* __restrict__ sums,
    const float* __restrict__ sumsq, const float* __restrict__ gamma,
    const float* __restrict__ beta) {
  const size_t i = (size_t)blockIdx.x * blockDim.x + threadIdx.x;
  const int col = (int)((i * 2) % C_OUT);
  const float inv_n = 1.0f / (float)N_PTS;
  float2 v = *((float2*)h + i);
  const float mu0 = sums[col] * inv_n;
  const float mu1 = sums[col + 1] * inv_n;
  const float var0 = fmaxf(sumsq[col] * inv_n - mu0 * mu0, 0.0f);
  const float var1 = fmaxf(sumsq[col + 1] * inv_n - mu1 * mu1, 0.0f);
  const float sc0 = gamma[col] * rsqrtf(var0 + BN_EPS);
  const float sc1 = gamma[col + 1] * rsqrtf(var1 + BN_EPS);
  v.x = fmaxf((v.x - mu0) * sc0 + beta[col], 0.0f);
  v.y = fmaxf((v.y - mu1) * sc1 + beta[col + 1], 0.0f);
  *((float2*)h + i) = v;
}

// ---------------------------------------------------------------------------
// FPS: single workgroup. pos (3*16K floats = 196KB) + dmin (64KB) live in the
// 320KB WGP LDS -> every one of the 8192 serial steps is LDS-only.
// Argmax with lowest-index tie break (matches jnp.argmax).
// ---------------------------------------------------------------------------
__global__ __launch_bounds__(1024) void fps_kernel(
    const float* __restrict__ pos, int* __restrict__ selected) {
  extern __shared__ float smem[];
  float* px   = smem;
  float* py   = smem + N_PTS;
  float* pz   = smem + 2 * N_PTS;
  float* dmin = smem + 3 * N_PTS;
  float* rv   = smem + 4 * N_PTS;          // 32 wave-best values
  int*   ri   = (int*)(rv + 32);           // 32 wave-best indices
  int*   bc   = ri + 32;                   // broadcast slot

  const int t = threadIdx.x;
  for (int p = t; p < N_PTS; p += 1024) {
    px[p] = pos[p * 3 + 0];
    py[p] = pos[p * 3 + 1];
    pz[p] = pos[p * 3 + 2];
    dmin[p] = __int_as_float(0x7f800000);  // +inf
  }
  __syncthreads();

  int last = 0;
  for (int it = 0; it < M_OUT; ++it) {
    if (t == 0) selected[last] = 1;
    const float lx = px[last], ly = py[last], lz = pz[last];
    float bestv = -1.0f;
    int   besti = N_PTS;
#pragma unroll 4
    for (int p = t; p < N_PTS; p += 1024) {
      const float dx = px[p] - lx, dy = py[p] - ly, dz = pz[p] - lz;
      const float d = dx * dx + dy * dy + dz * dz;
      const float dm = fminf(dmin[p], d);
      dmin[p] = dm;
      if (dm > bestv || (dm == bestv && p < besti)) { bestv = dm; besti = p; }
    }
    // wave32 argmax reduce
#pragma unroll
    for (int off = 16; off > 0; off >>= 1) {
      const float ov = __shfl_down(bestv, off, 32);
      const int   oi = __shfl_down(besti, off, 32);
      if (ov > bestv || (ov == bestv && oi < besti)) { bestv = ov; besti = oi; }
    }
    if ((t & 31) == 0) { rv[t >> 5] = bestv; ri[t >> 5] = besti; }
    __syncthreads();
    if (t < 32) {
      bestv = rv[t]; besti = ri[t];
#pragma unroll
      for (int off = 16; off > 0; off >>= 1) {
        const float ov = __shfl_down(bestv, off, 32);
        const int   oi = __shfl_down(besti, off, 32);
        if (ov > bestv || (ov == bestv && oi < besti)) { bestv = ov; besti = oi; }
      }
      if (t == 0) *bc = besti;
    }
    __syncthreads();
    last = *bc;
  }
}

// ---------------------------------------------------------------------------
// Compaction of selected flags -> sorted sample indices (ascending by
// construction). Block-wide Hillis-Steele scan over 1024 chunk counts.
// ---------------------------------------------------------------------------
__global__ __launch_bounds__(1024) void compact_kernel(
    const int* __restrict__ selected, int* __restrict__ idx_out) {
  __shared__ int s[1024];
  const int t = threadIdx.x;
  const int base = t * (N_PTS / 1024);  // 16 contiguous flags per thread
  int flags[16];
  int c = 0;
#pragma unroll
  for (int i = 0; i < 16; ++i) { flags[i] = selected[base + i]; c += flags[i]; }
  s[t] = c;
  __syncthreads();
  for (int off = 1; off < 1024; off <<= 1) {
    const int v = (t >= off) ? s[t - off] : 0;
    __syncthreads();
    s[t] += v;
    __syncthreads();
  }
  int w = s[t] - c;  // exclusive prefix
#pragma unroll
  for (int i = 0; i < 16; ++i)
    if (flags[i]) idx_out[w++] = base + i;
}

// ---------------------------------------------------------------------------
// Scatter-max over edges. h >= 0 after ReLU, so fp32 order == u32 order ->
// native u32 atomicMax (no CAS loop). 2 edges per 256-thread block,
// float4 of h[src] per thread.
// ---------------------------------------------------------------------------
__global__ __launch_bounds__(256) void scatter_max_kernel(
    const float* __restrict__ h, float* __restrict__ pooled,
    const int* __restrict__ edge_index) {
  const int e = blockIdx.x * 2 + (threadIdx.x >> 7);
  const int c = (threadIdx.x & 127) * 4;
  const int src = edge_index[e];
  const int dst = edge_index[E_EDGES + e];
  const float4 v = *(const float4*)(h + (size_t)src * C_OUT + c);
  unsigned* o = (unsigned*)(pooled + (size_t)dst * C_OUT + c);
  atomicMax(o + 0, __float_as_uint(v.x));
  atomicMax(o + 1, __float_as_uint(v.y));
  atomicMax(o + 2, __float_as_uint(v.z));
  atomicMax(o + 3, __float_as_uint(v.w));
}

// ---------------------------------------------------------------------------
// Gather final outputs: [M*512 pooled | M*3 pos | M batch(int bits)]
// ---------------------------------------------------------------------------
__global__ __launch_bounds__(256) void gather_kernel(
    const float* __restrict__ pooled, const float* __restrict__ pos,
    const int* __restrict__ batch, const int* __restrict__ idx,
    float* __restrict__ out) {
  const int i = blockIdx.x;
  const int t = threadIdx.x;
  const int p = idx[i];
  const float2* sp = (const float2*)(pooled + (size_t)p * C_OUT);
  float2* dp = (float2*)(out + (size_t)i * C_OUT);
  dp[t] = sp[t];
  float* out_pos = out + (size_t)M_OUT * C_OUT;
  if (t < 3) out_pos[i * 3 + t] = pos[p * 3 + t];
  if (t == 0) {
    int* out_batch = (int*)(out + (size_t)M_OUT * C_OUT + (size_t)M_OUT * 3);
    out_batch[i] = batch[p];
  }
}

// ---------------------------------------------------------------------------
extern "C" void kernel_launch(void* const* d_in, const int* in_sizes, int n_in,
                              void* d_out, int out_size, void* d_ws,
                              size_t ws_size, hipStream_t stream) {
  const float* x          = (const float*)d_in[0];
  const float* pos        = (const float*)d_in[1];
  const int*   batch      = (const int*)d_in[2];
  const int*   edge_index = (const int*)d_in[3];
  const float* W          = (const float*)d_in[4];
  const float* bias       = (const float*)d_in[5];
  const float* gamma      = (const float*)d_in[6];
  const float* beta       = (const float*)d_in[7];
  float* out = (float*)d_out;

  // workspace carve
  float* h        = (float*)d_ws;                 // N*C floats (32MB)
  float* pooled   = h + (size_t)N_PTS * C_OUT;    // N*C floats (32MB)
  float* sums     = pooled + (size_t)N_PTS * C_OUT;
  float* sumsq    = sums + C_OUT;
  int*   selected = (int*)(sumsq + C_OUT);        // N ints
  int*   idx      = selected + N_PTS;             // M ints

  init_kernel<<<(N_PTS + 255) / 256, 256, 0, stream>>>(sums, sumsq, selected);

  dim3 ggrid(N_PTS / 128, C_OUT / 64);
  gemm_wmma_kernel<<<ggrid, 256, 0, stream>>>(x, W, bias, h);

  bn_stats_kernel<<<N_PTS / 64, 256, 0, stream>>>(h, sums, sumsq);
  bn_apply_kernel<<<(size_t)N_PTS * C_OUT / 2 / 256, 256, 0, stream>>>(
      h, sums, sumsq, gamma, beta);

  const size_t fps_smem =
      (size_t)4 * N_PTS * sizeof(float) + 32 * sizeof(float) +
      32 * sizeof(int) + sizeof(int);
  hipFuncSetAttribute((const void*)fps_kernel,
                      hipFuncAttributeMaxDynamicSharedMemorySize,
                      (int)fps_smem);
  fps_kernel<<<1, 1024, fps_smem, stream>>>(pos, selected);

  compact_kernel<<<1, 1024, 0, stream>>>(selected, idx);

  hipMemcpyAsync(pooled, h, (size_t)N_PTS * C_OUT * sizeof(float),
                 hipMemcpyDeviceToDevice, stream);

  scatter_max_kernel<<<E_EDGES / 2, 256, 0, stream>>>(h, pooled, edge_index);

  gather_kernel<<<M_OUT, 256, 0, stream>>>(pooled, pos, batch, idx, out);
}